// RankingSet_53850299957682
// MI455X (gfx1250) — compile-verified
//
#include <hip/hip_runtime.h>
#include <hip/hip_bf16.h>

typedef float v2f __attribute__((ext_vector_type(2)));
typedef float v8f __attribute__((ext_vector_type(8)));

#define EPS_N 1e-12f
#define RTOL 1e-5f
#define ATOL 1e-8f

#define KC 192
#define LDSROW (KC + 4)   // stride 196 dwords == 4 (mod 64): conflict-free fragments

// ---------------------------------------------------------------- row norms
__global__ void rownorm_kernel(const float* __restrict__ data,
                               float* __restrict__ rnorm, int D) {
    __shared__ float red[256];
    const int row = blockIdx.x;
    const float* p = data + (size_t)row * D;
    float s = 0.f;
    for (int j = threadIdx.x; j < D; j += 256) { float v = p[j]; s += v * v; }
    red[threadIdx.x] = s;
    __syncthreads();
    for (int off = 128; off > 0; off >>= 1) {
        if (threadIdx.x < off) red[threadIdx.x] += red[threadIdx.x + off];
        __syncthreads();
    }
    if (threadIdx.x == 0) rnorm[row] = fmaxf(sqrtf(red[0]), EPS_N);
}

// ------------------------------------- normalize queries + per-query thresh
__global__ void qnorm_kernel(const float* __restrict__ q,
                             const float* __restrict__ t,
                             float* __restrict__ qn,
                             float* __restrict__ thresh, int D) {
    __shared__ float rq[256], rt[256], rqt[256];
    const int b = blockIdx.x;
    const float* qp = q + (size_t)b * D;
    const float* tp = t + (size_t)b * D;
    float sq = 0.f, st = 0.f, sqt = 0.f;
    for (int j = threadIdx.x; j < D; j += 256) {
        float qv = qp[j], tv = tp[j];
        sq += qv * qv; st += tv * tv; sqt += qv * tv;
    }
    rq[threadIdx.x] = sq; rt[threadIdx.x] = st; rqt[threadIdx.x] = sqt;
    __syncthreads();
    for (int off = 128; off > 0; off >>= 1) {
        if (threadIdx.x < off) {
            rq[threadIdx.x]  += rq[threadIdx.x + off];
            rt[threadIdx.x]  += rt[threadIdx.x + off];
            rqt[threadIdx.x] += rqt[threadIdx.x + off];
        }
        __syncthreads();
    }
    const float inq = 1.f / fmaxf(sqrtf(rq[0]), EPS_N);
    const float int_ = 1.f / fmaxf(sqrtf(rt[0]), EPS_N);
    for (int j = threadIdx.x; j < D; j += 256)
        qn[(size_t)b * D + j] = qp[j] * inq;
    if (threadIdx.x == 0) thresh[b] = rqt[0] * inq * int_;
}

// ------------------------------------------------ WMMA GEMM + rank counting
// Block = 8 waves. Wave w: rows [blk*128 + 16w, +16), all 128 queries.
// A fragment (fp32 16x16x4): lane<16 -> K{0,1}, lane>=16 -> K{2,3}; M = lane%16.
__global__ __launch_bounds__(256)
void rank_gemm_kernel(const float* __restrict__ data,
                      const float* __restrict__ qn,
                      const float* __restrict__ rnorm,
                      const float* __restrict__ thresh,
                      int* __restrict__ counts, int Nrows, int D) {
    __shared__ float ldsq[128 * LDSROW];   // 100,352 B (3 WGs per 320 KB WGP)

    const int tid  = threadIdx.x;
    const int lane = tid & 31;
    const int wave = tid >> 5;
    const int hi   = lane >> 4;     // K-half selector
    const int l15  = lane & 15;     // M (for A) / N (for B,C)
    const int rowBase = blockIdx.x * 128 + wave * 16;
    const int arow = min(rowBase + l15, Nrows - 1);
    const float* aptr = data + (size_t)arow * D + 2 * hi;

    v8f acc[8];
#pragma unroll
    for (int t = 0; t < 8; ++t)
#pragma unroll
        for (int e = 0; e < 8; ++e) acc[t][e] = 0.f;

    const int nchunks = D / KC;
    for (int c = 0; c < nchunks; ++c) {
        const int kb = c * KC;
        __syncthreads();
        // Cooperative stage of qn[0..127][kb, kb+KC) into LDS as float4s.
        const int nf4 = KC / 4;                         // float4 per row
        for (int idx = tid; idx < 128 * nf4; idx += 256) {
            const int r  = idx / nf4;
            const int c4 = idx - r * nf4;
            const float4 v = *(const float4*)(qn + (size_t)r * D + kb + c4 * 4);
            *(float4*)(&ldsq[r * LDSROW + c4 * 4]) = v;
        }
        __syncthreads();
        if (c + 1 < nchunks)                 // global_prefetch_b8 of next A chunk
            __builtin_prefetch(aptr + kb + KC, 0, 1);
#pragma unroll
        for (int s = 0; s < KC / 4; ++s) {
            const v2f a = *(const v2f*)(aptr + kb + s * 4);
            // Hoist all 8 B fragments: lets the backend clause the ds_loads and
            // use partial DScnt waits, then run 8 WMMAs back-to-back.
            v2f b[8];
#pragma unroll
            for (int t = 0; t < 8; ++t)
                b[t] = *(const v2f*)(&ldsq[(t * 16 + l15) * LDSROW + s * 4 + 2 * hi]);
#pragma unroll
            for (int t = 0; t < 8; ++t)
                acc[t] = __builtin_amdgcn_wmma_f32_16x16x4_f32(
                    /*neg_a=*/false, a, /*neg_b=*/false, b[t],
                    /*c_mod=*/(short)0, acc[t],
                    /*reuse_a=*/false, /*reuse_b=*/false);
        }
    }

    // Epilogue: C lane layout -> M = v + 8*hi, N = l15 (per tile t: N = 16t+l15)
    float invr[8];
    const int rowlim = Nrows - rowBase - 8 * hi;   // element v valid iff v < rowlim
#pragma unroll
    for (int v = 0; v < 8; ++v) {
        const int r = rowBase + 8 * hi + v;
        invr[v] = 1.0f / rnorm[min(r, Nrows - 1)];
    }
#pragma unroll
    for (int t = 0; t < 8; ++t) {
        const float th  = thresh[t * 16 + l15];
        const float tol = ATOL + RTOL * fabsf(th);
        int cnt = 0;
#pragma unroll
        for (int v = 0; v < 8; ++v) {
            const float p = acc[t][v] * invr[v];
            const bool g = (p >= th) || (fabsf(p - th) <= tol);
            cnt += (g && (v < rowlim)) ? 1 : 0;
        }
        cnt += __shfl_xor(cnt, 16, 32);                // fold lane <-> lane^16
        if (hi == 0) atomicAdd(&counts[t * 16 + l15], cnt);
    }
}

// ----------------------------------------------------------------- finalize
__global__ void finalize_kernel(const int* __restrict__ counts,
                                float* __restrict__ out, int Q) {
    const int i = threadIdx.x + blockIdx.x * blockDim.x;
    if (i < Q) out[i] = (float)(counts[i] - 1);
}

// ------------------------------------------------------------------- launch
extern "C" void kernel_launch(void* const* d_in, const int* in_sizes, int n_in,
                              void* d_out, int out_size, void* d_ws, size_t ws_size,
                              hipStream_t stream) {
    const float* data    = (const float*)d_in[0];
    const float* queries = (const float*)d_in[1];
    const float* truths  = (const float*)d_in[2];
    float* out = (float*)d_out;

    const int Q = out_size;                 // 128
    const int D = in_sizes[1] / Q;          // 6144
    const int Nrows = in_sizes[0] / D;      // 50000

    char* ws = (char*)d_ws;
    size_t off = 0;
    float* rnorm = (float*)(ws + off); off += (((size_t)Nrows * 4) + 255) & ~(size_t)255;
    float* qn    = (float*)(ws + off); off += (((size_t)Q * D * 4) + 255) & ~(size_t)255;
    float* thr   = (float*)(ws + off); off += (((size_t)Q * 4) + 255) & ~(size_t)255;
    int*   cnts  = (int*)  (ws + off); off += (((size_t)Q * 4) + 255) & ~(size_t)255;
    (void)ws_size; (void)n_in;

    hipMemsetAsync(cnts, 0, Q * sizeof(int), stream);
    rownorm_kernel<<<Nrows, 256, 0, stream>>>(data, rnorm, D);
    qnorm_kernel<<<Q, 256, 0, stream>>>(queries, truths, qn, thr, D);
    const int nblk = (Nrows + 127) / 128;
    rank_gemm_kernel<<<nblk, 256, 0, stream>>>(data, qn, rnorm, thr, cnts, Nrows, D);
    finalize_kernel<<<1, 256, 0, stream>>>(cnts, out, Q);
}